// CubeAttention_26972394619720
// MI455X (gfx1250) — compile-verified
//
#include <hip/hip_runtime.h>
#include <math.h>

// CubeAttention for MI455X (gfx1250): all contractions via V_WMMA_F32_16X16X4_F32.
// K-pairs interleaved so every WMMA fragment is one aligned 8-byte load.
// Coronal conv stages its input plane with GLOBAL_LOAD_ASYNC_TO_LDS_B128
// (ASYNCcnt-tracked); long-latency fragment loads are software-pipelined
// (depth 2) so their waits land behind the previous step's WMMA burst.
//
// Workspace (floats):
//   xg    [2][64][64][64][64]  channel-major input           (33,554,432)
//   tans  [b][c][w1][w][d]     transverse conv output        (33,554,432)
//   cor   [b][c][w1][h][w]     coronal conv output           (33,554,432)
//   fprod [b][c][h][w][d]      shortcut * f                  (33,554,432)
//   attn  [b][c][h][u][d]                                    (33,554,432)
//   wtp   [9][64][128][2]      packed transverse weights     (147,456)
//   wcp   [9][64][128][2]      packed coronal weights        (147,456)
// Total ws = 672,268,288 bytes.

typedef __attribute__((ext_vector_type(2))) float v2f;
typedef __attribute__((ext_vector_type(4))) float v4f;
typedef __attribute__((ext_vector_type(8))) float v8f;

#define NVOX 262144            // 64*64*64 voxels per channel
#define CHVOX (64 * 262144)    // per-batch (64 channels) volume

static __device__ __forceinline__ v8f wmma4(v2f a, v2f b, v8f c) {
  // D(16x16,f32) = A(16x4,f32) * B(4x16,f32) + C
  return __builtin_amdgcn_wmma_f32_16x16x4_f32(false, a, false, b, (short)0, c,
                                               false, false);
}

static __device__ __forceinline__ float gelu_exact(float x) {
  return 0.5f * x * (1.0f + erff(x * 0.70710678118654752f));
}

// ---------------------------------------------------------------- pack x -> xg
__global__ __launch_bounds__(256) void k_pack_x(const float* __restrict__ x,
                                                float* __restrict__ xg) {
  __shared__ float t[64][65];
  int b  = blockIdx.x >> 12;
  int n0 = (blockIdx.x & 4095) << 6;
  const float* xp = x + ((size_t)b * NVOX + n0) * 64;
  for (int e = threadIdx.x; e < 4096; e += 256) {
    int r = e >> 6, c = e & 63;
    t[r][c] = xp[(size_t)r * 64 + c];
  }
  __syncthreads();
  float* xo = xg + (size_t)b * CHVOX + n0;
  for (int e = threadIdx.x; e < 4096; e += 256) {
    int c = e >> 6, r = e & 63;
    xo[(size_t)c * NVOX + r] = t[r][c];
  }
}

// --------------------------------- pack conv weights: [s][k/2][oc][2] (K-pairs)
__global__ void k_pack_wt(const float* __restrict__ wt, float* __restrict__ wp) {
  int idx = blockIdx.x * 256 + threadIdx.x;
  if (idx >= 9 * 128 * 128) return;
  int oc = idx & 127, k = (idx >> 7) & 127, s = idx >> 14;
  int i = k >> 6, h = k & 63;
  int opos = ((s * 64 + (k >> 1)) * 128 + oc) * 2 + (k & 1);
  wp[opos] = wt[(size_t)((oc * 2 + i) * 64 + h) * 9 + s];
}
__global__ void k_pack_wc(const float* __restrict__ wc, float* __restrict__ wp) {
  int idx = blockIdx.x * 256 + threadIdx.x;
  if (idx >= 9 * 128 * 128) return;
  int oc = idx & 127, k = (idx >> 7) & 127, s = idx >> 14;
  int i = k >> 6, d = k & 63;
  int opos = ((s * 64 + (k >> 1)) * 128 + oc) * 2 + (k & 1);
  wp[opos] = wc[((size_t)(oc * 2 + i) * 9 + s) * 64 + d];
}

// --------------------------------------------------------- transverse conv GEMM
// Block = (b'=b*32+cg, w). plane stores K-pairs: [k/2][d+1][2].
__global__ __launch_bounds__(256) void k_conv_t(const float* __restrict__ xg,
                                                const float* __restrict__ wp,
                                                const float* __restrict__ bias,
                                                float* __restrict__ tans) {
  __shared__ float plane[64][66][2];
  int w  = blockIdx.x & 63;
  int bp = blockIdx.x >> 6;
  int b = bp >> 5, cg = bp & 31;
  const float* xin = xg + (size_t)b * CHVOX + (size_t)(cg * 2) * NVOX;
  int wave = threadIdx.x >> 5, lane = threadIdx.x & 31;
  int lm = lane & 15, lh = lane >> 4;

  {                                       // zero d-padding columns (dp=0,65) once
    int row = threadIdx.x >> 1;
    int edge = (threadIdx.x & 1) ? 65 : 0;
    plane[row >> 1][edge][row & 1] = 0.f;
  }

  v8f acc[4] = {};
  int aidx = wave * 16 + lm;
  for (int kh = 0; kh < 3; ++kh) {
    int wq = w + kh - 1;
    if (wq < 0 || wq > 63) continue;      // all-zero plane: contributes nothing
    __syncthreads();
    // interior: 128 rows x 16 chunks(16B) = 2048 = 8 per thread, b128 loads
#pragma unroll
    for (int k = 0; k < 8; ++k) {
      int cidx = threadIdx.x + (k << 8);
      int row = cidx >> 4, part = cidx & 15;
      v4f val = *(const v4f*)(xin + (size_t)row * 4096 + wq * 64 + (part << 2));
      int dp = (part << 2) + 1;
#pragma unroll
      for (int j = 0; j < 4; ++j)
        plane[row >> 1][dp + j][row & 1] = val[j];
    }
    __syncthreads();
    for (int kw = 0; kw < 3; ++kw) {
      int s = kh * 3 + kw;
      const float* wpp = wp + s * 16384;  // [k/2][oc][2]
      if (s + 1 < 9)                       // prefetch next 64KB weight slab
        __builtin_prefetch(wpp + 16384 + threadIdx.x * 64, 0, 1);
      v2f a_cur = *(const v2f*)(wpp + (lh * 128 + aidx) * 2);
      for (int k0 = 0; k0 < 128; k0 += 4) {
        int k4 = (k0 + 4) & 127;          // wraps: harmless redundant load
        v2f a_nxt = *(const v2f*)(wpp + (((k4 >> 1) + lh) * 128 + aidx) * 2);
        int kp = (k0 >> 1) + lh;
#pragma unroll
        for (int nt = 0; nt < 4; ++nt) {
          v2f bb = *(const v2f*)&plane[kp][nt * 16 + lm + kw][0];
          acc[nt] = wmma4(a_cur, bb, acc[nt]);
        }
        a_cur = a_nxt;
      }
    }
  }
  // oc = wave*16 + m; tans[b][c=2cg+oc/64][w1=oc%64][w][d]
#pragma unroll
  for (int e = 0; e < 8; ++e) {
    int m = e + 8 * lh;
    int oc = wave * 16 + m;
    int c = cg * 2 + (oc >> 6);
    int w1 = oc & 63;
    float bv = bias[oc];
    size_t base = ((((size_t)b * 64 + c) * 64 + w1) * 64 + w) * 64;
#pragma unroll
    for (int nt = 0; nt < 4; ++nt)
      tans[base + nt * 16 + lm] = acc[nt][e] + bv;
  }
}

// ----------------------------------------------------------- coronal conv GEMM
// Block = (b', h). plane[i][w+1][d] (stride 68 => every row 16B-aligned).
// Interior staged via GLOBAL_LOAD_ASYNC_TO_LDS_B128; edges zeroed once.
__global__ __launch_bounds__(256) void k_conv_c(const float* __restrict__ xg,
                                                const float* __restrict__ wp,
                                                const float* __restrict__ bias,
                                                float* __restrict__ cor) {
  __shared__ float plane[2][66][68];
  int h  = blockIdx.x & 63;
  int bp = blockIdx.x >> 6;
  int b = bp >> 5, cg = bp & 31;
  const float* xin = xg + (size_t)b * CHVOX + (size_t)(cg * 2) * NVOX;
  int wave = threadIdx.x >> 5, lane = threadIdx.x & 31;
  int lm = lane & 15, lh = lane >> 4;

  // zero the w-padding edge columns (wpi = 0 and 65) once
  for (int e = threadIdx.x; e < 2 * 2 * 68; e += 256) {
    int d = e & 63;
    int t = e >> 6;
    plane[t >> 1][(t & 1) ? 65 : 0][d] = 0.f;
    if (d < 4) plane[t >> 1][(t & 1) ? 65 : 0][64 + d] = 0.f;
  }

  uint32_t ldsbase = (uint32_t)(uintptr_t)&plane[0][0][0];
  v8f acc[4] = {};
  int aidx = wave * 16 + lm;
  for (int kh = 0; kh < 3; ++kh) {
    int hq = h + kh - 1;
    if (hq < 0 || hq > 63) continue;      // all-zero plane: contributes nothing
    __syncthreads();                      // prior-iteration LDS reads complete
    // 128 interior rows x 16 chunks(16B) = 2048 chunks = 8 per thread (uniform)
#pragma unroll
    for (int k = 0; k < 8; ++k) {
      int cidx = threadIdx.x + (k << 8);
      int row  = cidx >> 4;               // 0..127
      int part = cidx & 15;               // 16B sub-chunk
      int i = row >> 6, wpi = (row & 63) + 1;
      uint32_t goff = (uint32_t)((i * NVOX + hq * 4096 + (wpi - 1) * 64 +
                                  (part << 2)) << 2);
      uint32_t loff = ldsbase +
                      (uint32_t)(((i * 66 + wpi) * 68 + (part << 2)) << 2);
      asm volatile("global_load_async_to_lds_b128 %0, %1, %2"
                   :: "v"(loff), "v"(goff), "s"(xin) : "memory");
    }
    asm volatile("s_wait_asynccnt 0x0" ::: "memory");
    __syncthreads();
    for (int kw = 0; kw < 3; ++kw) {
      int s = kh * 3 + kw;
      const float* wpp = wp + s * 16384;
      if (s + 1 < 9)
        __builtin_prefetch(wpp + 16384 + threadIdx.x * 64, 0, 1);
      v2f a_cur = *(const v2f*)(wpp + (lh * 128 + aidx) * 2);
      for (int k0 = 0; k0 < 128; k0 += 4) {
        int k4 = (k0 + 4) & 127;
        v2f a_nxt = *(const v2f*)(wpp + (((k4 >> 1) + lh) * 128 + aidx) * 2);
        int kk = k0 + 2 * lh;
        int i = kk >> 6, d = kk & 63;     // kk even -> pair (d, d+1) same i
#pragma unroll
        for (int nt = 0; nt < 4; ++nt) {
          int wcol = nt * 16 + lm + kw;
          v2f bb = *(const v2f*)&plane[i][wcol][d];
          acc[nt] = wmma4(a_cur, bb, acc[nt]);
        }
        a_cur = a_nxt;
      }
    }
  }
  // cor[b][c][w1][h][w]
#pragma unroll
  for (int e = 0; e < 8; ++e) {
    int m = e + 8 * lh;
    int oc = wave * 16 + m;
    int c = cg * 2 + (oc >> 6);
    int w1 = oc & 63;
    float bv = bias[oc];
    size_t base = ((((size_t)b * 64 + c) * 64 + w1) * 64 + h) * 64;
#pragma unroll
    for (int nt = 0; nt < 4; ++nt)
      cor[base + nt * 16 + lm] = acc[nt][e] + bv;
  }
}

// ------------------------------------------------ f = sum_w1 cor*tans, *shortcut
__global__ __launch_bounds__(128) void k_f(const float* __restrict__ cor,
                                           const float* __restrict__ tans,
                                           const float* __restrict__ xg,
                                           float* __restrict__ fprod) {
  __shared__ float sc[32][66][2];   // [w1/2][h][2]
  __shared__ float st[32][66][2];   // [w1/2][d][2]
  int w = blockIdx.x & 63;
  int c = (blockIdx.x >> 6) & 63;
  int b = blockIdx.x >> 12;
  size_t cbase = ((size_t)b * 64 + c) * NVOX;
  const float* corp = cor + cbase;
  const float* tanp = tans + cbase;
  for (int e = threadIdx.x; e < 4096; e += 128) {       // strided gather
    int w1 = e >> 6, q = e & 63;
    sc[w1 >> 1][q][w1 & 1] = corp[(size_t)w1 * 4096 + q * 64 + w];
  }
  for (int e4 = threadIdx.x * 4; e4 < 4096; e4 += 512) { // contiguous rows
    int w1 = e4 >> 6, q = e4 & 63;
    v4f vv = *(const v4f*)(tanp + (size_t)w1 * 4096 + w * 64 + q);
#pragma unroll
    for (int j = 0; j < 4; ++j) st[w1 >> 1][q + j][w1 & 1] = vv[j];
  }
  __syncthreads();
  int wave = threadIdx.x >> 5, lane = threadIdx.x & 31;
  int lm = lane & 15, lh = lane >> 4;
  v8f acc[4] = {};
  for (int k0 = 0; k0 < 64; k0 += 4) {
    int kp = (k0 >> 1) + lh;
    v2f a = *(const v2f*)&sc[kp][wave * 16 + lm][0];
#pragma unroll
    for (int nt = 0; nt < 4; ++nt) {
      v2f bb = *(const v2f*)&st[kp][nt * 16 + lm][0];
      acc[nt] = wmma4(a, bb, acc[nt]);
    }
  }
#pragma unroll
  for (int e = 0; e < 8; ++e) {
    int hrow = wave * 16 + e + 8 * lh;
    size_t rb = cbase + (size_t)hrow * 4096 + (size_t)w * 64;
#pragma unroll
    for (int nt = 0; nt < 4; ++nt) {
      int d = nt * 16 + lm;
      fprod[rb + d] = acc[nt][e] * xg[rb + d];
    }
  }
}

// ------------------------------------------------------ attn = sum_w cor*tans
__global__ __launch_bounds__(128) void k_attn(const float* __restrict__ cor,
                                              const float* __restrict__ tans,
                                              float* __restrict__ attn) {
  __shared__ float scT[32][66][2];  // [w/2][h][2] (transposed on store)
  __shared__ float st[32][66][2];   // [w/2][d][2]
  int u = blockIdx.x & 63;
  int c = (blockIdx.x >> 6) & 63;
  int b = blockIdx.x >> 12;
  size_t cbase = ((size_t)b * 64 + c) * NVOX;
  const float* corp = cor + cbase + (size_t)u * 4096;   // [h][w]
  const float* tanp = tans + cbase + (size_t)u * 4096;  // [w][d]
  for (int e4 = threadIdx.x * 4; e4 < 4096; e4 += 512) {
    int r = e4 >> 6, q = e4 & 63;
    v4f a4 = *(const v4f*)(corp + e4);   // r = h, q..q+3 = w (K dim)
#pragma unroll
    for (int j = 0; j < 4; ++j) scT[(q + j) >> 1][r][(q + j) & 1] = a4[j];
    v4f b4 = *(const v4f*)(tanp + e4);   // r = w (K dim), q..q+3 = d
#pragma unroll
    for (int j = 0; j < 4; ++j) st[r >> 1][q + j][r & 1] = b4[j];
  }
  __syncthreads();
  int wave = threadIdx.x >> 5, lane = threadIdx.x & 31;
  int lm = lane & 15, lh = lane >> 4;
  v8f acc[4] = {};
  for (int k0 = 0; k0 < 64; k0 += 4) {
    int kp = (k0 >> 1) + lh;
    v2f a = *(const v2f*)&scT[kp][wave * 16 + lm][0];
#pragma unroll
    for (int nt = 0; nt < 4; ++nt) {
      v2f bb = *(const v2f*)&st[kp][nt * 16 + lm][0];
      acc[nt] = wmma4(a, bb, acc[nt]);
    }
  }
#pragma unroll
  for (int e = 0; e < 8; ++e) {
    int hrow = wave * 16 + e + 8 * lh;
    size_t rb = cbase + (size_t)hrow * 4096 + (size_t)u * 64;
#pragma unroll
    for (int nt = 0; nt < 4; ++nt)
      attn[rb + nt * 16 + lm] = acc[nt][e];
  }
}

// -------------------------------------- 1x1 map GEMM + bias + exact GELU + NHWC
__global__ __launch_bounds__(256) void k_map(const float* __restrict__ fprod,
                                             const float* __restrict__ attnb,
                                             const float* __restrict__ wmap,
                                             const float* __restrict__ bmap,
                                             float* __restrict__ out) {
  __shared__ float wm[64][64][2];  // [k/2][cout][2]
  for (int e4 = threadIdx.x * 4; e4 < 8192; e4 += 1024) {
    v4f vv = *(const v4f*)(wmap + e4);     // 4 consecutive k, same cout m
    int m = e4 >> 7, k = e4 & 127;
    v2f lo = {vv.x, vv.y}, hi = {vv.z, vv.w};
    *(v2f*)&wm[(k >> 1) + 0][m][0] = lo;
    *(v2f*)&wm[(k >> 1) + 1][m][0] = hi;
  }
  __syncthreads();
  int b  = blockIdx.x >> 11;
  int v0 = (blockIdx.x & 2047) << 7;
  int wave = threadIdx.x >> 5, lane = threadIdx.x & 31;
  int lm = lane & 15, lh = lane >> 4;
  int v = v0 + wave * 16 + lm;
  const float* fp = fprod + (size_t)b * CHVOX;
  const float* ap = attnb + (size_t)b * CHVOX;

  auto bload = [&](int k0) -> v2f {
    int kk = k0 + 2 * lh;                  // even; pair never straddles k=64
    const float* src = (kk < 64) ? (fp + (size_t)kk * NVOX)
                                 : (ap + (size_t)(kk - 64) * NVOX);
    v2f r;
    r.x = src[v];
    r.y = src[(size_t)NVOX + v];
    return r;
  };

  v8f acc[4] = {};
  v2f b_cur = bload(0);
  for (int k0 = 0; k0 < 128; k0 += 4) {
    v2f b_nxt = bload((k0 + 4) & 127);     // wraps: harmless redundant load
    int kk = k0 + 2 * lh;
#pragma unroll
    for (int mt = 0; mt < 4; ++mt) {
      v2f a = *(const v2f*)&wm[kk >> 1][mt * 16 + lm][0];
      acc[mt] = wmma4(a, b_cur, acc[mt]);
    }
    b_cur = b_nxt;
  }
  // lane's 8 acc elements are consecutive cout values -> two b128 stores per mt
#pragma unroll
  for (int mt = 0; mt < 4; ++mt) {
    int cb = mt * 16 + 8 * lh;
    v4f g0, g1;
#pragma unroll
    for (int e = 0; e < 4; ++e)
      g0[e] = gelu_exact(acc[mt][e] + bmap[cb + e]);
#pragma unroll
    for (int e = 4; e < 8; ++e)
      g1[e - 4] = gelu_exact(acc[mt][e] + bmap[cb + e]);
    float* op = out + ((size_t)b * NVOX + v) * 64 + cb;
    *(v4f*)op = g0;
    *(v4f*)(op + 4) = g1;
  }
}

extern "C" void kernel_launch(void* const* d_in, const int* in_sizes, int n_in,
                              void* d_out, int out_size, void* d_ws, size_t ws_size,
                              hipStream_t stream) {
  (void)in_sizes; (void)n_in; (void)out_size; (void)ws_size;
  const float* x     = (const float*)d_in[0];
  const float* w_t   = (const float*)d_in[1];
  const float* b_t   = (const float*)d_in[2];
  const float* w_c   = (const float*)d_in[3];
  const float* b_c   = (const float*)d_in[4];
  const float* w_map = (const float*)d_in[5];
  const float* b_map = (const float*)d_in[6];
  float* out = (float*)d_out;

  float* ws    = (float*)d_ws;
  float* xg    = ws;
  float* tans  = xg    + 33554432;
  float* cor   = tans  + 33554432;
  float* fprod = cor   + 33554432;
  float* attn  = fprod + 33554432;
  float* wtp   = attn  + 33554432;
  float* wcp   = wtp   + 147456;

  k_pack_x <<<8192, 256, 0, stream>>>(x, xg);
  k_pack_wt<<<576,  256, 0, stream>>>(w_t, wtp);
  k_pack_wc<<<576,  256, 0, stream>>>(w_c, wcp);
  k_conv_t <<<4096, 256, 0, stream>>>(xg, wtp, b_t, tans);
  k_conv_c <<<4096, 256, 0, stream>>>(xg, wcp, b_c, cor);
  k_f      <<<8192, 128, 0, stream>>>(cor, tans, xg, fprod);
  k_attn   <<<8192, 128, 0, stream>>>(cor, tans, attn);
  k_map    <<<4096, 256, 0, stream>>>(fprod, attn, w_map, b_map, out);
}